// HierarchicalMemorySystem_67894843015428
// MI455X (gfx1250) — compile-verified
//
#include <hip/hip_runtime.h>
#include <math.h>

typedef float v2f __attribute__((ext_vector_type(2)));
typedef float v8f __attribute__((ext_vector_type(8)));

#define WAVE      32
#define NWAVES    8            // 2 M-tiles x 4 N-groups
#define BLOCK     (WAVE * NWAVES)
#define K_DIM     256
#define TILE_M    16
#define ROWS_BLK  32           // 2 M-tiles per block
#define XS_STRIDE 260          // 260 % 64 == 4 -> conflict-free column reads

// ---------------------------------------------------------------------------
// Per-prototype squared norms: one wave per prototype row (256 dims / 32 lanes).
// ---------------------------------------------------------------------------
__global__ __launch_bounds__(256) void wnorm_kernel(const float* __restrict__ w,
                                                    int n_rows,
                                                    float* __restrict__ out) {
    const int wid  = threadIdx.x / WAVE;
    const int lane = threadIdx.x % WAVE;
    const int row  = blockIdx.x * 8 + wid;
    if (row >= n_rows) return;
    const float* wr = w + (size_t)row * K_DIM;
    float s = 0.0f;
#pragma unroll
    for (int j = 0; j < K_DIM / WAVE; ++j) {
        float v = wr[lane + j * WAVE];
        s = fmaf(v, v, s);
    }
#pragma unroll
    for (int mask = 16; mask >= 1; mask >>= 1)
        s += __shfl_xor(s, mask, WAVE);
    if (lane == 0) out[row] = s;
}

// ---------------------------------------------------------------------------
// One block = 32 batch rows (2 M-tiles). 8 waves = (mw in {0,1}) x (nw in 0..3).
// Each wave sweeps prototype-column tile PAIRS (2*nw, 2*nw+1) stepping by 8,
// driving two independent V_WMMA_F32_16X16X4_F32 accumulation chains that
// share one A fragment per K-step. Track running min of (||w||^2 - 2*x.w),
// argmin-reduce across lanes (N) and the 4 N-group waves, then recompute the
// exact Euclidean distance to the BMU (matches the reference's second pass).
// ---------------------------------------------------------------------------
__global__ __launch_bounds__(BLOCK) void som_level_kernel(
    const float* __restrict__ x, const float* __restrict__ w,
    const float* __restrict__ wnorm, int n_tiles, int side,
    float* __restrict__ out_c, float* __restrict__ out_q) {

    __shared__ float xs[ROWS_BLK * XS_STRIDE];
    __shared__ float smin[4][2][TILE_M];
    __shared__ int   sidx[4][2][TILE_M];
    __shared__ int   sbmu[ROWS_BLK];

    const int row0 = blockIdx.x * ROWS_BLK;
    const int tid  = threadIdx.x;
    const int wid  = tid / WAVE;
    const int lane = tid % WAVE;
    const int mw   = wid & 1;    // which 16-row M-tile this wave owns
    const int nw   = wid >> 1;   // N interleave group (0..3)
    const int nloc = lane & 15;  // N (or M for A-frag) position within tile
    const int hi   = lane >> 4;  // lane half selects K sub-block per ISA layout

    // Stage the 32x256 x tile into LDS (padded stride).
    for (int i = tid; i < ROWS_BLK * K_DIM; i += BLOCK) {
        const int m = i >> 8;
        const int k = i & (K_DIM - 1);
        xs[m * XS_STRIDE + k] = x[(size_t)(row0 + m) * K_DIM + k];
    }
    __syncthreads();

    // Running best (value, global n) per accumulator register r.
    // C/D layout: VGPR r, lanes 0-15 -> M=r ; lanes 16-31 -> M=r+8.
    float bv[8];
    int   bi[8];
#pragma unroll
    for (int r = 0; r < 8; ++r) { bv[r] = 3.4e38f; bi[r] = 0; }

    // A fragment base: lane m = nloc of this wave's M-tile, K pair by lane half.
    const float* ap_base = &xs[(mw * TILE_M + nloc) * XS_STRIDE + 2 * hi];

    for (int t = nw * 2; t < n_tiles; t += 2 * NWAVES / 2) {  // step 8
        const int ca = t * 16;          // first tile columns
        const int cb = ca + 16;         // paired tile columns
        const float* wpa = w + ((size_t)(ca + nloc)) * K_DIM + 2 * hi;
        const float* wpb = wpa + (size_t)16 * K_DIM;
        if (t + 8 < n_tiles) {
            __builtin_prefetch(wpa + (size_t)8 * 16 * K_DIM, 0, 0);
            __builtin_prefetch(wpb + (size_t)8 * 16 * K_DIM, 0, 0);
        }

        v8f acc0 = {};
        v8f acc1 = {};
#pragma unroll 8
        for (int k0 = 0; k0 < K_DIM; k0 += 4) {
            // A 16x4 f32: V0 = K0 (lanes 0-15) / K2 (lanes 16-31); V1 = K1/K3.
            v2f a, b0, b1;
            a.x  = ap_base[k0];
            a.y  = ap_base[k0 + 1];
            // B 4x16 f32 (= w^T tile): same K striping, lanes carry N.
            b0.x = wpa[k0];
            b0.y = wpa[k0 + 1];
            b1.x = wpb[k0];
            b1.y = wpb[k0 + 1];
            acc0 = __builtin_amdgcn_wmma_f32_16x16x4_f32(
                false, a, false, b0, (short)0, acc0, false, false);
            acc1 = __builtin_amdgcn_wmma_f32_16x16x4_f32(
                false, a, false, b1, (short)0, acc1, false, false);
        }

        const float wna = wnorm[ca + nloc];
        const float wnb = wnorm[cb + nloc];
        const int   na  = ca + nloc;
        const int   nb  = cb + nloc;
#pragma unroll
        for (int r = 0; r < 8; ++r) {
            const float da = fmaf(-2.0f, acc0[r], wna);
            const float db = fmaf(-2.0f, acc1[r], wnb);
            if (da < bv[r] || (da == bv[r] && na < bi[r])) { bv[r] = da; bi[r] = na; }
            if (db < bv[r] || (db == bv[r] && nb < bi[r])) { bv[r] = db; bi[r] = nb; }
        }
    }

    // Cross-lane argmin over the 16 N-lanes of each half-wave.
#pragma unroll
    for (int r = 0; r < 8; ++r) {
        float v  = bv[r];
        int   ix = bi[r];
#pragma unroll
        for (int mask = 8; mask >= 1; mask >>= 1) {
            const float ov = __shfl_xor(v, mask, WAVE);
            const int   oi = __shfl_xor(ix, mask, WAVE);
            if (ov < v || (ov == v && oi < ix)) { v = ov; ix = oi; }
        }
        if (nloc == 0) {
            const int m = r + 8 * hi;
            smin[nw][mw][m] = v;
            sidx[nw][mw][m] = ix;
        }
    }
    __syncthreads();

    // Cross-wave reduce over the 4 N-groups + coordinate write.
    if (tid < ROWS_BLK) {
        const int mt = tid >> 4;
        const int m  = tid & 15;
        float v  = smin[0][mt][m];
        int   ix = sidx[0][mt][m];
#pragma unroll
        for (int g = 1; g < 4; ++g) {
            const float ov = smin[g][mt][m];
            const int   oi = sidx[g][mt][m];
            if (ov < v || (ov == v && oi < ix)) { v = ov; ix = oi; }
        }
        sbmu[tid] = ix;
        out_c[(size_t)(row0 + tid) * 2 + 0] = (float)(ix / side);
        out_c[(size_t)(row0 + tid) * 2 + 1] = (float)(ix % side);
    }
    __syncthreads();

    // Exact quantization error: q = ||x - w[bmu]||, 4 rows per wave.
#pragma unroll
    for (int j = 0; j < ROWS_BLK / NWAVES; ++j) {
        const int m   = wid * (ROWS_BLK / NWAVES) + j;
        const int bmu = sbmu[m];
        const float* wr = w + (size_t)bmu * K_DIM;
        float s = 0.0f;
#pragma unroll
        for (int jj = 0; jj < K_DIM / WAVE; ++jj) {
            const int k = lane + jj * WAVE;
            const float d = xs[m * XS_STRIDE + k] - wr[k];
            s = fmaf(d, d, s);
        }
#pragma unroll
        for (int mask = 16; mask >= 1; mask >>= 1)
            s += __shfl_xor(s, mask, WAVE);
        if (lane == 0) out_q[row0 + m] = sqrtf(s);
    }
}

extern "C" void kernel_launch(void* const* d_in, const int* in_sizes, int n_in,
                              void* d_out, int out_size, void* d_ws, size_t ws_size,
                              hipStream_t stream) {
    (void)in_sizes; (void)n_in; (void)out_size; (void)ws_size;
    const float* x  = (const float*)d_in[0];   // 8192 x 256
    const float* w1 = (const float*)d_in[1];   //  256 x 256
    const float* w2 = (const float*)d_in[2];   // 1024 x 256
    const float* w3 = (const float*)d_in[3];   // 4096 x 256

    float* out = (float*)d_out;
    float* c1 = out;                    // 8192*2
    float* c2 = out + 16384;            // 8192*2
    float* c3 = out + 32768;            // 8192*2
    float* q1 = out + 49152;            // 8192
    float* q2 = q1 + 8192;
    float* q3 = q2 + 8192;

    float* wn1 = (float*)d_ws;          //  256
    float* wn2 = wn1 + 256;             // 1024
    float* wn3 = wn2 + 1024;            // 4096

    wnorm_kernel<<<256 / 8,  256, 0, stream>>>(w1, 256,  wn1);
    wnorm_kernel<<<1024 / 8, 256, 0, stream>>>(w2, 1024, wn2);
    wnorm_kernel<<<4096 / 8, 256, 0, stream>>>(w3, 4096, wn3);

    const int grid = 8192 / ROWS_BLK;  // 256 blocks, 8 wave32 each
    som_level_kernel<<<grid, BLOCK, 0, stream>>>(x, w1, wn1, 256 / 16,  16, c1, q1);
    som_level_kernel<<<grid, BLOCK, 0, stream>>>(x, w2, wn2, 1024 / 16, 32, c2, q2);
    som_level_kernel<<<grid, BLOCK, 0, stream>>>(x, w3, wn3, 4096 / 16, 64, c3, q3);
}